// MambaBlock_28106265985451
// MI455X (gfx1250) — compile-verified
//
#include <hip/hip_runtime.h>
#include <hip/hip_bf16.h>

#define BB 16
#define CC 192
#define HH 48
#define WW 48
#define LL 2304            // H*W
#define DI 288
#define NS 16
#define KK4 4
#define BL (BB*LL)         // 36864

typedef float v2f __attribute__((ext_vector_type(2)));
typedef float v8f __attribute__((ext_vector_type(8)));

__device__ __forceinline__ v8f wmma4(v2f a, v2f b, v8f c) {
  // D = A(16x4,f32) * B(4x16,f32) + C(16x16,f32)  -> v_wmma_f32_16x16x4_f32
  return __builtin_amdgcn_wmma_f32_16x16x4_f32(false, a, false, b, (short)0, c, false, false);
}

// ---------------------------------------------------------------------------
// K1: xz[b*L+l][0..575] = sum_c x[b][c][l] * W_in[c][j]
// M = B*L = 36864, N = 576, K = 192. One wave per 16x16 tile.
__global__ void k_inproj(const float* __restrict__ x, const float* __restrict__ Wi,
                         float* __restrict__ xz) {
  int lane = threadIdx.x & 31;
  int wave = (blockIdx.x * blockDim.x + threadIdx.x) >> 5;   // 82944 waves
  int mt = wave / 36, nt = wave % 36;
  int m0 = mt * 16, n0 = nt * 16;
  int lr = lane & 15, hb = lane >> 4;
  int b = m0 / LL, l0 = m0 % LL;
  const float* xb = x + (size_t)b * CC * LL;
  v8f acc = {0.f,0.f,0.f,0.f,0.f,0.f,0.f,0.f};
  for (int kk = 0; kk < CC; kk += 4) {
    int ka = kk + hb * 2;
    v2f a, bf;
    a.x  = xb[(size_t)(ka + 0) * LL + l0 + lr];
    a.y  = xb[(size_t)(ka + 1) * LL + l0 + lr];
    bf.x = Wi[(size_t)(ka + 0) * 576 + n0 + lr];
    bf.y = Wi[(size_t)(ka + 1) * 576 + n0 + lr];
    acc = wmma4(a, bf, acc);
  }
  float* out = xz + (size_t)(m0 + 8 * hb) * 576 + n0 + lr;
#pragma unroll
  for (int r = 0; r < 8; ++r) out[(size_t)r * 576] = acc[r];
}

// ---------------------------------------------------------------------------
// K2: depthwise 3x3 conv + bias + SiLU, write both scan orders (l-major, d fastest)
__global__ void k_conv(const float* __restrict__ xz, const float* __restrict__ cw,
                       const float* __restrict__ cb,
                       float* __restrict__ us0, float* __restrict__ us1) {
  int d = threadIdx.x;            // 288
  int bl = blockIdx.x;            // 36864
  int b = bl / LL, l = bl % LL;
  int h = l / WW, w = l % WW;
  float acc = cb[d];
#pragma unroll
  for (int dh = -1; dh <= 1; ++dh)
#pragma unroll
    for (int dw = -1; dw <= 1; ++dw) {
      int hh = h + dh, ww = w + dw;
      if (hh >= 0 && hh < HH && ww >= 0 && ww < WW) {
        acc += cw[d * 9 + (dh + 1) * 3 + (dw + 1)] *
               xz[((size_t)b * LL + hh * WW + ww) * 576 + d];
      }
    }
  float v = acc / (1.f + __expf(-acc));
  us0[((size_t)b * LL + h * WW + w) * DI + d] = v;
  us1[((size_t)b * LL + w * HH + h) * DI + d] = v;
}

// ---------------------------------------------------------------------------
// K3: x_dbl[bk][l][c] = sum_d us_{k&1}[b][l][d] * xpw[k][c][d]
// per (b,k): M = L = 2304, N = 48, K = 288
__global__ void k_xdbl(const float* __restrict__ us0, const float* __restrict__ us1,
                       const float* __restrict__ xpw, float* __restrict__ xdbl) {
  int lane = threadIdx.x & 31;
  int wave = (blockIdx.x * blockDim.x + threadIdx.x) >> 5;   // 27648 waves
  int bk = wave / 432, t = wave % 432;
  int mt = t / 3, nt = t % 3;
  int m0 = mt * 16, n0 = nt * 16;
  int lr = lane & 15, hb = lane >> 4;
  int b = bk >> 2, k = bk & 3;
  const float* us = ((k & 1) ? us1 : us0) + (size_t)b * LL * DI;
  const float* wv = xpw + (size_t)k * 48 * DI;
  v8f acc = {0.f,0.f,0.f,0.f,0.f,0.f,0.f,0.f};
  for (int kk = 0; kk < DI; kk += 4) {
    int ka = kk + hb * 2;
    v2f a, bf;
    a.x  = us[(size_t)(m0 + lr) * DI + ka + 0];
    a.y  = us[(size_t)(m0 + lr) * DI + ka + 1];
    bf.x = wv[(size_t)(n0 + lr) * DI + ka + 0];
    bf.y = wv[(size_t)(n0 + lr) * DI + ka + 1];
    acc = wmma4(a, bf, acc);
  }
  float* out = xdbl + ((size_t)bk * LL + m0 + 8 * hb) * 48 + n0 + lr;
#pragma unroll
  for (int r = 0; r < 8; ++r) out[(size_t)r * 48] = acc[r];
}

// ---------------------------------------------------------------------------
// K4: dt[bk][l][d] = softplus( sum_r xdbl[bk][l][r] * dtw[k][d][r] + dtb[k][d] )
// per (b,k): M = L, N = 288, K = 16
__global__ void k_dt(const float* __restrict__ xdbl, const float* __restrict__ dtw,
                     const float* __restrict__ dtb, float* __restrict__ dt) {
  int lane = threadIdx.x & 31;
  int wave = (blockIdx.x * blockDim.x + threadIdx.x) >> 5;   // 165888 waves
  int bk = wave / 2592, t = wave % 2592;
  int mt = t / 18, nt = t % 18;
  int m0 = mt * 16, n0 = nt * 16;
  int lr = lane & 15, hb = lane >> 4;
  int k = bk & 3;
  const float* xa = xdbl + (size_t)bk * LL * 48;
  const float* wv = dtw + (size_t)k * DI * NS;
  v8f acc = {0.f,0.f,0.f,0.f,0.f,0.f,0.f,0.f};
#pragma unroll
  for (int kk = 0; kk < NS; kk += 4) {
    int ka = kk + hb * 2;
    v2f a, bf;
    a.x  = xa[(size_t)(m0 + lr) * 48 + ka + 0];
    a.y  = xa[(size_t)(m0 + lr) * 48 + ka + 1];
    bf.x = wv[(size_t)(n0 + lr) * NS + ka + 0];
    bf.y = wv[(size_t)(n0 + lr) * NS + ka + 1];
    acc = wmma4(a, bf, acc);
  }
  float bias = dtb[k * DI + n0 + lr];
  float* out = dt + ((size_t)bk * LL + m0 + 8 * hb) * DI + n0 + lr;
#pragma unroll
  for (int r = 0; r < 8; ++r) {
    float v = acc[r] + bias;
    out[(size_t)r * DI] = (v > 20.f) ? v : __logf(1.f + __expf(v));
  }
}

// ---------------------------------------------------------------------------
// K5: selective scan. Block = (b,k,half), 576 threads = 144 d x 4 state-quads.
// ys stored at base index (so cross-merge is a plain add).
__global__ void k_scan(const float* __restrict__ us0, const float* __restrict__ us1,
                       const float* __restrict__ xdbl, const float* __restrict__ dt,
                       const float* __restrict__ A_log, const float* __restrict__ Ds,
                       float* __restrict__ ys) {
  int blk = blockIdx.x;                 // 128
  int half = blk & 1, bk = blk >> 1;
  int b = bk >> 2, k = bk & 3;
  int t = threadIdx.x;
  int s = t & 3;
  int d = half * 144 + (t >> 2);
  const float* us  = ((k & 1) ? us1 : us0) + (size_t)b * LL * DI;
  const float* dtp = dt   + (size_t)bk * LL * DI;
  const float* xd  = xdbl + (size_t)bk * LL * 48;
  float*       yp_ = ys   + (size_t)bk * LL * DI;
  float A2[4], hst[4] = {0.f, 0.f, 0.f, 0.f};
#pragma unroll
  for (int j = 0; j < 4; ++j)
    A2[j] = -__expf(A_log[((size_t)k * DI + d) * NS + s * 4 + j]);
  float Dk = Ds[k * DI + d];
  bool fwd = (k < 2);
  __shared__ float bc[32][32];          // rows of (B|C), 16+16 floats
  for (int c0 = 0; c0 < LL; c0 += 32) {
    int lb0 = fwd ? c0 : (LL - c0 - 32);
    __syncthreads();
    for (int idx = t; idx < 1024; idx += 576) {
      int row = idx >> 5, c = idx & 31;
      bc[row][c] = xd[(size_t)(lb0 + row) * 48 + 16 + c];
    }
    __syncthreads();
    for (int i = 0; i < 32; ++i) {
      int rl = fwd ? i : (31 - i);
      int lb = lb0 + rl;
      float u  = us [(size_t)lb * DI + d];
      float dv = dtp[(size_t)lb * DI + d];
      float du = dv * u;
      float yq = 0.f;
#pragma unroll
      for (int j = 0; j < 4; ++j) {
        float dA = __expf(dv * A2[j]);
        hst[j] = fmaf(dA, hst[j], du * bc[rl][s * 4 + j]);
        yq = fmaf(hst[j], bc[rl][16 + s * 4 + j], yq);
      }
      yq += __shfl_xor(yq, 1);
      yq += __shfl_xor(yq, 2);
      if (s == 0) yp_[(size_t)lb * DI + d] = fmaf(Dk, u, yq);
    }
  }
}

// ---------------------------------------------------------------------------
// K6: cross-merge + LayerNorm(288) + SiLU(z) gate. Block per (b,l), 288 threads.
__global__ void k_merge(const float* __restrict__ ys, const float* __restrict__ xz,
                        const float* __restrict__ lg, const float* __restrict__ lb,
                        float* __restrict__ yg) {
  int bl = blockIdx.x;
  int b = bl / LL, l = bl % LL;
  int h = l / WW, w = l % WW;
  int lt = w * HH + h;
  int d = threadIdx.x;
  size_t base = (size_t)b * KK4 * LL * DI;
  float y = ys[base + ((size_t)0 * LL + l ) * DI + d]
          + ys[base + ((size_t)2 * LL + l ) * DI + d]
          + ys[base + ((size_t)1 * LL + lt) * DI + d]
          + ys[base + ((size_t)3 * LL + lt) * DI + d];
  float s = y, q = y * y;
#pragma unroll
  for (int o = 16; o >= 1; o >>= 1) { s += __shfl_xor(s, o); q += __shfl_xor(q, o); }
  __shared__ float wsum[9], wsq[9], mv[2];
  int wid = threadIdx.x >> 5;
  if ((threadIdx.x & 31) == 0) { wsum[wid] = s; wsq[wid] = q; }
  __syncthreads();
  if (threadIdx.x == 0) {
    float S = 0.f, Q = 0.f;
    for (int i = 0; i < 9; ++i) { S += wsum[i]; Q += wsq[i]; }
    float mu = S * (1.f / DI);
    mv[0] = mu; mv[1] = Q * (1.f / DI) - mu * mu;
  }
  __syncthreads();
  float yn = (y - mv[0]) * rsqrtf(mv[1] + 1e-5f) * lg[d] + lb[d];
  float z = xz[((size_t)bl) * 576 + DI + d];
  yg[(size_t)bl * DI + d] = yn * (z / (1.f + __expf(-z)));
}

// ---------------------------------------------------------------------------
// K7: o[b*L+l][c] = sum_d yg[b*L+l][d] * W_out[d][c].  M=36864, N=192, K=288.
__global__ void k_outproj(const float* __restrict__ yg, const float* __restrict__ Wo,
                          float* __restrict__ o) {
  int lane = threadIdx.x & 31;
  int wave = (blockIdx.x * blockDim.x + threadIdx.x) >> 5;   // 27648 waves
  int mt = wave / 12, nt = wave % 12;
  int m0 = mt * 16, n0 = nt * 16;
  int lr = lane & 15, hb = lane >> 4;
  v8f acc = {0.f,0.f,0.f,0.f,0.f,0.f,0.f,0.f};
  for (int kk = 0; kk < DI; kk += 4) {
    int ka = kk + hb * 2;
    v2f a, bf;
    a.x  = yg[(size_t)(m0 + lr) * DI + ka + 0];
    a.y  = yg[(size_t)(m0 + lr) * DI + ka + 1];
    bf.x = Wo[(size_t)(ka + 0) * CC + n0 + lr];
    bf.y = Wo[(size_t)(ka + 1) * CC + n0 + lr];
    acc = wmma4(a, bf, acc);
  }
  float* out = o + (size_t)(m0 + 8 * hb) * CC + n0 + lr;
#pragma unroll
  for (int r = 0; r < 8; ++r) out[(size_t)r * CC] = acc[r];
}

// ---------------------------------------------------------------------------
// K8a: zero BN accumulators. K8b: per-channel sum/sumsq with atomics.
__global__ void k_zero(float* p, int n) {
  int i = blockIdx.x * blockDim.x + threadIdx.x;
  if (i < n) p[i] = 0.f;
}
__global__ void k_bnstats(const float* __restrict__ o, float* __restrict__ st) {
  int c = threadIdx.x;                  // 192
  int r0 = blockIdx.x * 144;            // 256 blocks
  float s = 0.f, q = 0.f;
  for (int i = 0; i < 144; ++i) {
    float v = o[(size_t)(r0 + i) * CC + c];
    s += v; q += v * v;
  }
  atomicAdd(&st[c], s);
  atomicAdd(&st[CC + c], q);
}

// ---------------------------------------------------------------------------
// K9: BN apply + NHWC -> NCHW transpose through padded LDS.
__global__ void k_bnapply(const float* __restrict__ o, const float* __restrict__ st,
                          const float* __restrict__ bg, const float* __restrict__ bb,
                          float* __restrict__ out) {
  __shared__ float tile[32 * 193];
  __shared__ float sc[CC], sh[CC];
  int b = blockIdx.x / 72;
  int l0 = (blockIdx.x % 72) * 32;
  for (int c = threadIdx.x; c < CC; c += blockDim.x) {
    float m = st[c] * (1.f / (float)BL);
    float v = st[CC + c] * (1.f / (float)BL) - m * m;
    float s = rsqrtf(v + 1e-5f) * bg[c];
    sc[c] = s; sh[c] = bb[c] - m * s;
  }
  __syncthreads();
  for (int idx = threadIdx.x; idx < 32 * CC; idx += blockDim.x) {
    int i = idx / CC, c = idx % CC;
    tile[i * 193 + c] = o[(size_t)(b * LL + l0 + i) * CC + c] * sc[c] + sh[c];
  }
  __syncthreads();
  for (int idx = threadIdx.x; idx < 32 * CC; idx += blockDim.x) {
    int c = idx / 32, i = idx % 32;
    out[(size_t)b * CC * LL + (size_t)c * LL + l0 + i] = tile[i * 193 + c];
  }
}

// ---------------------------------------------------------------------------
extern "C" void kernel_launch(void* const* d_in, const int* in_sizes, int n_in,
                              void* d_out, int out_size, void* d_ws, size_t ws_size,
                              hipStream_t stream) {
  const float* x    = (const float*)d_in[0];
  const float* Wi   = (const float*)d_in[1];
  const float* cw   = (const float*)d_in[2];
  const float* cb   = (const float*)d_in[3];
  const float* xpw  = (const float*)d_in[4];
  const float* dtw  = (const float*)d_in[5];
  const float* dtb  = (const float*)d_in[6];
  const float* Alog = (const float*)d_in[7];
  const float* Dsk  = (const float*)d_in[8];
  const float* lg   = (const float*)d_in[9];
  const float* lb   = (const float*)d_in[10];
  const float* Wo   = (const float*)d_in[11];
  const float* bg   = (const float*)d_in[12];
  const float* bb   = (const float*)d_in[13];
  float* out = (float*)d_out;

  float* ws = (float*)d_ws;
  const size_t OFF_XZ   = 0;                           // 21,233,664
  const size_t OFF_US0  = OFF_XZ  + (size_t)BL * 576;  // 10,616,832
  const size_t OFF_US1  = OFF_US0 + (size_t)BL * DI;
  const size_t OFF_XDBL = OFF_US1 + (size_t)BL * DI;   //  7,077,888
  const size_t OFF_DT   = OFF_XDBL + (size_t)BB * KK4 * LL * 48;   // 42,467,328
  const size_t OFF_YS   = OFF_DT  + (size_t)BB * KK4 * LL * DI;    // 42,467,328
  const size_t OFF_ST   = OFF_YS  + (size_t)BB * KK4 * LL * DI;    // 384
  float* xz   = ws + OFF_XZ;
  float* us0  = ws + OFF_US0;
  float* us1  = ws + OFF_US1;
  float* xdbl = ws + OFF_XDBL;
  float* dt   = ws + OFF_DT;
  float* ys   = ws + OFF_YS;
  float* yg   = ws + OFF_DT;   // alias: dt dead after scan
  float* o    = ws + OFF_YS;   // alias: ys dead after merge
  float* st   = ws + OFF_ST;

  // K1: in-projection, 82944 waves
  k_inproj<<<82944 / 8, 256, 0, stream>>>(x, Wi, xz);
  // K2: depthwise conv + SiLU + cross-scan bases
  k_conv<<<BL, DI, 0, stream>>>(xz, cw, cb, us0, us1);
  // K3: x_dbl projection, 27648 waves
  k_xdbl<<<27648 / 8, 256, 0, stream>>>(us0, us1, xpw, xdbl);
  // K4: dt projection + softplus, 165888 waves
  k_dt<<<165888 / 8, 256, 0, stream>>>(xdbl, dtw, dtb, dt);
  // K5: selective scan
  k_scan<<<BB * KK4 * 2, 576, 0, stream>>>(us0, us1, xdbl, dt, Alog, Dsk, ys);
  // K6: merge + LayerNorm + gate
  k_merge<<<BL, DI, 0, stream>>>(ys, xz, lg, lb, yg);
  // K7: out-projection, 27648 waves
  k_outproj<<<27648 / 8, 256, 0, stream>>>(yg, Wo, o);
  // K8: BatchNorm statistics
  k_zero<<<1, 384, 0, stream>>>(st, 384);
  k_bnstats<<<256, CC, 0, stream>>>(o, st);
  // K9: BatchNorm apply + NCHW transpose
  k_bnapply<<<BB * (LL / 32), 256, 0, stream>>>(o, st, bg, bb, out);
}